// MHCN_Encoder_35003983462546
// MI455X (gfx1250) — compile-verified
//
#include <hip/hip_runtime.h>
#include <math.h>

typedef float v2f __attribute__((ext_vector_type(2)));
typedef float v8f __attribute__((ext_vector_type(8)));

// ---------------------------------------------------------------------------
// vec[d] = sum_j att_mat[d][j] * att_agg[j]   (collapses channel attention to
// a per-row dot product: sum(att_agg * (e@att_mat),axis=1) == e @ (att_mat@att_agg^T))
// ---------------------------------------------------------------------------
__global__ void attvec_kernel(const float* __restrict__ att_mat,
                              const float* __restrict__ att_agg,
                              float* __restrict__ vec) {
  int d = threadIdx.x;  // 64 threads
  float s = 0.f;
#pragma unroll
  for (int j = 0; j < 64; ++j) s += att_mat[d * 64 + j] * att_agg[j];
  vec[d] = s;
}

// ---------------------------------------------------------------------------
// out = em * sigmoid(em @ W + b)   for em:[nrows,64], W:[64,64], b:[64]
// WMMA f32 16x16x4 path: each wave computes one 16-row x 64-col tile.
// W and b staged in LDS (16KB of the 320KB WGP pool).
// ---------------------------------------------------------------------------
__global__ void gate_kernel(const float* __restrict__ em,
                            const float* __restrict__ W,
                            const float* __restrict__ b,
                            float* __restrict__ out, int nrows) {
  __shared__ float sW[64 * 64];
  __shared__ float sb[64];
  for (int i = threadIdx.x; i < 64 * 64; i += 256) sW[i] = W[i];
  if (threadIdx.x < 64) sb[threadIdx.x] = b[threadIdx.x];
  __syncthreads();

  const int wave   = threadIdx.x >> 5;
  const int lane   = threadIdx.x & 31;
  const int lane16 = lane & 15;
  const int khalf  = (lane >> 4) << 1;  // lanes 0-15 hold K0/K1, lanes 16-31 hold K2/K3
  const int rowBase = (blockIdx.x * 8 + wave) * 16;
  if (rowBase + 16 > nrows) return;  // nrows is a multiple of 16 here

  v8f c0 = {}, c1 = {}, c2 = {}, c3 = {};
  const float* arow = em + (size_t)(rowBase + lane16) * 64;

  for (int k = 0; k < 64; k += 4) {
    v2f a;
    a.x = arow[k + khalf];
    a.y = arow[k + khalf + 1];
    const float* w0 = &sW[(k + khalf) * 64 + lane16];
    const float* w1 = &sW[(k + khalf + 1) * 64 + lane16];
    v2f b0; b0.x = w0[0];  b0.y = w1[0];
    v2f b1; b1.x = w0[16]; b1.y = w1[16];
    v2f b2; b2.x = w0[32]; b2.y = w1[32];
    v2f b3; b3.x = w0[48]; b3.y = w1[48];
    c0 = __builtin_amdgcn_wmma_f32_16x16x4_f32(false, a, false, b0, (short)0, c0, false, false);
    c1 = __builtin_amdgcn_wmma_f32_16x16x4_f32(false, a, false, b1, (short)0, c1, false, false);
    c2 = __builtin_amdgcn_wmma_f32_16x16x4_f32(false, a, false, b2, (short)0, c2, false, false);
    c3 = __builtin_amdgcn_wmma_f32_16x16x4_f32(false, a, false, b3, (short)0, c3, false, false);
  }

  // C/D layout: VGPR i holds M = i (lanes 0-15) or M = i+8 (lanes 16-31), N = lane%16 + 16*j
  const int mOff = (lane >> 4) << 3;
  v8f cs[4] = {c0, c1, c2, c3};
#pragma unroll
  for (int j = 0; j < 4; ++j) {
    const int coln = j * 16 + lane16;
#pragma unroll
    for (int i = 0; i < 8; ++i) {
      const int row = rowBase + mOff + i;
      const float acc = cs[j][i] + sb[coln];
      const float g = 1.0f / (1.0f + __expf(-acc));
      out[(size_t)row * 64 + coln] = em[(size_t)row * 64 + coln] * g;
    }
  }
}

// ---------------------------------------------------------------------------
// Channel attention (via precomputed vec) + combine with "simple" channel:
//   out = amix * softmax-mix(e0,e1,e2) + bsimple * simple
// One wave32 per row, 2 elements per lane, shfl_xor reductions.
// ---------------------------------------------------------------------------
__global__ void mix_kernel(const float* __restrict__ e0, const float* __restrict__ e1,
                           const float* __restrict__ e2, const float* __restrict__ simple,
                           const float* __restrict__ vec, float* __restrict__ out,
                           int nrows, float amix, float bsimple) {
  const int wave = threadIdx.x >> 5;
  const int lane = threadIdx.x & 31;
  const int row = blockIdx.x * 8 + wave;
  if (row >= nrows) return;
  const size_t base = (size_t)row * 64;
  const float v0 = vec[lane], v1 = vec[lane + 32];
  const float x00 = e0[base + lane], x01 = e0[base + lane + 32];
  const float x10 = e1[base + lane], x11 = e1[base + lane + 32];
  const float x20 = e2[base + lane], x21 = e2[base + lane + 32];
  float w0 = x00 * v0 + x01 * v1;
  float w1 = x10 * v0 + x11 * v1;
  float w2 = x20 * v0 + x21 * v1;
#pragma unroll
  for (int off = 16; off > 0; off >>= 1) {
    w0 += __shfl_xor(w0, off, 32);
    w1 += __shfl_xor(w1, off, 32);
    w2 += __shfl_xor(w2, off, 32);
  }
  const float m = fmaxf(w0, fmaxf(w1, w2));
  const float ex0 = __expf(w0 - m), ex1 = __expf(w1 - m), ex2 = __expf(w2 - m);
  const float inv = 1.0f / (ex0 + ex1 + ex2);
  const float s0 = ex0 * inv, s1 = ex1 * inv, s2 = ex2 * inv;
  out[base + lane]      = amix * (s0 * x00 + s1 * x10 + s2 * x20) + bsimple * simple[base + lane];
  out[base + lane + 32] = amix * (s0 * x01 + s1 * x11 + s2 * x21) + bsimple * simple[base + lane + 32];
}

// ---------------------------------------------------------------------------
// COO SpMM scatter: out[row[e], :] += val[e] * x[col[e], :]
// 64 lanes cover D (coalesced 256B gather + coalesced f32 atomic scatter).
// Edge streams (row/col/val, ~38MB/pass, zero reuse) are loaded non-temporal
// so they pass through without evicting the gather/scatter targets
// (x/out, 25.6MB each, ~32x reuse) from the 192MB L2.
// ---------------------------------------------------------------------------
__global__ void spmm_kernel(const int* __restrict__ rows, const int* __restrict__ cols,
                            const float* __restrict__ vals, const float* __restrict__ x,
                            float* __restrict__ out, int nnz) {
  const int d = threadIdx.x & 63;
  const int e = blockIdx.x * 4 + (threadIdx.x >> 6);
  if (e >= nnz) return;
  const int r   = __builtin_nontemporal_load(rows + e);
  const int c   = __builtin_nontemporal_load(cols + e);
  const float v = __builtin_nontemporal_load(vals + e);
  const float contrib = v * x[(size_t)c * 64 + d];
  (void)__hip_atomic_fetch_add(&out[(size_t)r * 64 + d], contrib,
                               __ATOMIC_RELAXED, __HIP_MEMORY_SCOPE_AGENT);
}

// ---------------------------------------------------------------------------
// acc[row,:] += x[row,:] / max(||x[row,:]||2, eps). One wave32 per row.
// ---------------------------------------------------------------------------
__global__ void l2acc_kernel(const float* __restrict__ x, float* __restrict__ acc, int nrows) {
  const int wave = threadIdx.x >> 5;
  const int lane = threadIdx.x & 31;
  const int row = blockIdx.x * 8 + wave;
  if (row >= nrows) return;
  const size_t base = (size_t)row * 64;
  const float x0 = x[base + lane], x1 = x[base + lane + 32];
  float ss = x0 * x0 + x1 * x1;
#pragma unroll
  for (int off = 16; off > 0; off >>= 1) ss += __shfl_xor(ss, off, 32);
  const float inv = 1.0f / fmaxf(sqrtf(ss), 1e-12f);
  acc[base + lane]      += x0 * inv;
  acc[base + lane + 32] += x1 * inv;
}

// ---------------------------------------------------------------------------
extern "C" void kernel_launch(void* const* d_in, const int* in_sizes, int n_in,
                              void* d_out, int out_size, void* d_ws, size_t ws_size,
                              hipStream_t stream) {
  const float* user_emb  = (const float*)d_in[0];
  const float* item_emb  = (const float*)d_in[1];
  const float* gating_w  = (const float*)d_in[2];
  const float* gating_b  = (const float*)d_in[3];
  const float* sgating_w = (const float*)d_in[4];
  const float* sgating_b = (const float*)d_in[5];
  const float* att_mat   = (const float*)d_in[6];
  const float* att_agg   = (const float*)d_in[7];
  const int*   hs_row = (const int*)d_in[8];
  const int*   hs_col = (const int*)d_in[9];
  const float* hs_val = (const float*)d_in[10];
  const int*   hj_row = (const int*)d_in[11];
  const int*   hj_col = (const int*)d_in[12];
  const float* hj_val = (const float*)d_in[13];
  const int*   hp_row = (const int*)d_in[14];
  const int*   hp_col = (const int*)d_in[15];
  const float* hp_val = (const float*)d_in[16];
  const int*   inter_row = (const int*)d_in[17];
  const int*   inter_col = (const int*)d_in[18];
  const float* inter_val = (const float*)d_in[19];

  const int NU   = in_sizes[0] / 64;   // 100000
  const int NI   = in_sizes[1] / 64;   // 50000
  const int nnzH = in_sizes[8];        // 3.2M
  const int nnzI = in_sizes[17];       // 3.0M
  const size_t U   = (size_t)NU * 64;
  const size_t Isz = (size_t)NI * 64;

  // workspace layout: 10*U + 2*I + 64 floats (~282 MB)
  float* ws = (float*)d_ws;
  float* uc0 = ws;           float* uc1 = ws + U;       float* uc2 = ws + 2 * U;
  float* usimple = ws + 3 * U;
  float* ac0 = ws + 4 * U;   float* ac1 = ws + 5 * U;   float* ac2 = ws + 6 * U;
  float* asimple = ws + 7 * U;
  float* mixed = ws + 8 * U;
  float* tmp   = ws + 9 * U;
  float* itA   = ws + 10 * U;
  float* itB   = itA + Isz;
  float* vec   = itB + Isz;

  float* out_final   = (float*)d_out;        // final_user [NU,64]
  float* out_accitem = out_final + U;        // acc_item   [NI,64]
  float* out_sg[3]   = { out_accitem + Isz,
                         out_accitem + Isz + U,
                         out_accitem + Isz + 2 * U };

  const dim3 B256(256);
  const int gateBlocks  = (NU / 16 + 7) / 8;
  const int rowBlocksU  = (NU + 7) / 8;
  const int rowBlocksI  = (NI + 7) / 8;
  const int spmmBlocksH = (nnzH + 3) / 4;
  const int spmmBlocksI = (nnzI + 3) / 4;

  // --- initial self-gatings (WMMA GEMMs) ---
  float* ug[4] = { uc0, uc1, uc2, usimple };
  for (int g = 0; g < 4; ++g)
    gate_kernel<<<gateBlocks, B256, 0, stream>>>(
        user_emb, gating_w + (size_t)g * 64 * 64, gating_b + (size_t)g * 64, ug[g], NU);

  hipMemcpyAsync(ac0, uc0, U * sizeof(float), hipMemcpyDeviceToDevice, stream);
  hipMemcpyAsync(ac1, uc1, U * sizeof(float), hipMemcpyDeviceToDevice, stream);
  hipMemcpyAsync(ac2, uc2, U * sizeof(float), hipMemcpyDeviceToDevice, stream);
  hipMemcpyAsync(asimple, usimple, U * sizeof(float), hipMemcpyDeviceToDevice, stream);
  hipMemcpyAsync(itA, item_emb, Isz * sizeof(float), hipMemcpyDeviceToDevice, stream);
  hipMemcpyAsync(out_accitem, item_emb, Isz * sizeof(float), hipMemcpyDeviceToDevice, stream);

  attvec_kernel<<<1, 64, 0, stream>>>(att_mat, att_agg, vec);  // constant across layers

  for (int layer = 0; layer < 2; ++layer) {
    // mixed = (channel_attention(u_c0,u_c1,u_c2) + u_simple) / 2   (uses pre-update values)
    mix_kernel<<<rowBlocksU, B256, 0, stream>>>(uc0, uc1, uc2, usimple, vec, mixed, NU, 0.5f, 0.5f);

    // u_c0 = spmm(hs, u_c0); acc_c0 += l2norm(u_c0)
    hipMemsetAsync(tmp, 0, U * sizeof(float), stream);
    spmm_kernel<<<spmmBlocksH, B256, 0, stream>>>(hs_row, hs_col, hs_val, uc0, tmp, nnzH);
    l2acc_kernel<<<rowBlocksU, B256, 0, stream>>>(tmp, ac0, NU);
    { float* t = uc0; uc0 = tmp; tmp = t; }

    // u_c1 = spmm(hj, u_c1); acc_c1 += l2norm(u_c1)
    hipMemsetAsync(tmp, 0, U * sizeof(float), stream);
    spmm_kernel<<<spmmBlocksH, B256, 0, stream>>>(hj_row, hj_col, hj_val, uc1, tmp, nnzH);
    l2acc_kernel<<<rowBlocksU, B256, 0, stream>>>(tmp, ac1, NU);
    { float* t = uc1; uc1 = tmp; tmp = t; }

    // u_c2 = spmm(hp, u_c2); acc_c2 += l2norm(u_c2)
    hipMemsetAsync(tmp, 0, U * sizeof(float), stream);
    spmm_kernel<<<spmmBlocksH, B256, 0, stream>>>(hp_row, hp_col, hp_val, uc2, tmp, nnzH);
    l2acc_kernel<<<rowBlocksU, B256, 0, stream>>>(tmp, ac2, NU);
    { float* t = uc2; uc2 = tmp; tmp = t; }

    // new_item = spmm(inter^T, mixed); acc_item += l2norm(new_item)
    hipMemsetAsync(itB, 0, Isz * sizeof(float), stream);
    spmm_kernel<<<spmmBlocksI, B256, 0, stream>>>(inter_col, inter_row, inter_val, mixed, itB, nnzI);
    l2acc_kernel<<<rowBlocksI, B256, 0, stream>>>(itB, out_accitem, NI);

    // u_simple = spmm(inter, it_old); acc_simple += l2norm(u_simple)
    hipMemsetAsync(tmp, 0, U * sizeof(float), stream);
    spmm_kernel<<<spmmBlocksI, B256, 0, stream>>>(inter_row, inter_col, inter_val, itA, tmp, nnzI);
    l2acc_kernel<<<rowBlocksU, B256, 0, stream>>>(tmp, asimple, NU);
    { float* t = usimple; usimple = tmp; tmp = t; }

    // it = new_item
    { float* t = itA; itA = itB; itB = t; }
  }

  // final_user = channel_attention(acc_c0,acc_c1,acc_c2) + acc_simple/2
  mix_kernel<<<rowBlocksU, B256, 0, stream>>>(ac0, ac1, ac2, asimple, vec, out_final, NU, 1.0f, 0.5f);

  // sg0..sg2 = self_gating(final_user, sgating_w[k], sgating_b[k])  (WMMA GEMMs)
  for (int g = 0; g < 3; ++g)
    gate_kernel<<<gateBlocks, B256, 0, stream>>>(
        out_final, sgating_w + (size_t)g * 64 * 64, sgating_b + (size_t)g * 64, out_sg[g], NU);
}